// Attention_65128884077225
// MI455X (gfx1250) — compile-verified
//
#include <hip/hip_runtime.h>

// ---------------------------------------------------------------------------
// Problem constants (B=4, S=2048, E=512, H=8). The reference collapses to:
//   xmean[b,:]  = mean_s x[b,s,:]                       (16 MB read)
//   vmean[b,:]  = xmean[b,:] @ qkv_w[:, 2E:3E] + qkv_b[2E:3E]
//   row[b,:]    = vmean[b,:] @ out_w + out_b
//   out[b,s,:]  = row[b,:]                              (16 MB write)
// Bandwidth bound: ~33 MB @ 23.3 TB/s ~= 1.4 us.
// ---------------------------------------------------------------------------

typedef float v2f __attribute__((ext_vector_type(2)));
typedef float v8f __attribute__((ext_vector_type(8)));

#define B_      4
#define S_      2048
#define E_      512
#define E4_     (E_ / 4)      // 128 float4 per row
#define SCHUNKS 64
#define SPER    (S_ / SCHUNKS) // 32 rows per chunk

// Phase 1: partial sums over S. grid = B_*SCHUNKS blocks, 128 threads.
// Each thread owns one float4 column, sums 32 consecutive s-rows.
__global__ void attn_partial_sum(const float4* __restrict__ x4,
                                 float4* __restrict__ partial) {
    const int b  = blockIdx.x >> 6;        // / SCHUNKS
    const int sc = blockIdx.x & (SCHUNKS - 1);
    const int e4 = threadIdx.x;            // 0..127
    float4 acc = make_float4(0.f, 0.f, 0.f, 0.f);
    int base = (b * S_ + sc * SPER) * E4_ + e4;
#pragma unroll 4
    for (int i = 0; i < SPER; ++i) {
        float4 v = x4[base + i * E4_];
        acc.x += v.x; acc.y += v.y; acc.z += v.z; acc.w += v.w;
    }
    partial[(sc * B_ + b) * E4_ + e4] = acc;
}

// Phase 2: combine 64 partials, scale by 1/S, write padded 16x512 A matrix
// (rows 4..15 zeroed). grid = 8 blocks x 256 threads = 2048 float4 slots.
__global__ void attn_finalize_mean(const float4* __restrict__ partial,
                                   float4* __restrict__ xmeanPad4) {
    const int idx = blockIdx.x * blockDim.x + threadIdx.x; // 0..2047
    const int row = idx >> 7;      // / E4_
    const int e4  = idx & (E4_ - 1);
    float4 r = make_float4(0.f, 0.f, 0.f, 0.f);
    if (row < B_) {
        for (int c = 0; c < SCHUNKS; ++c) {
            float4 v = partial[(c * B_ + row) * E4_ + e4];
            r.x += v.x; r.y += v.y; r.z += v.z; r.w += v.w;
        }
        const float inv = 1.0f / (float)S_;
        r.x *= inv; r.y *= inv; r.z *= inv; r.w *= inv;
    }
    xmeanPad4[row * E4_ + e4] = r;
}

// Small GEMM via fp32 WMMA: Out[16,512] = A[16,512] @ W[512, wPitch][:, colOff:colOff+512] + bias.
// One wave (32 threads) per 16-column tile; grid = 32 blocks.
// K loop in steps of 4 using V_WMMA_F32_16X16X4_F32 (full fp32 precision).
__global__ void attn_gemm16_wmma(const float* __restrict__ A,    // 16 x E_, pitch E_
                                 const float* __restrict__ W,    // E_ x wPitch
                                 int wPitch, int colOff,
                                 const float* __restrict__ bias, // pre-offset; index [n]
                                 float* __restrict__ Out) {      // 16 x E_, pitch E_
    const int n0   = blockIdx.x * 16;
    const int lane = threadIdx.x;      // 0..31, single wave32
    const int half = lane >> 4;        // 0: K={0,1}, 1: K={2,3}
    const int l    = lane & 15;        // M row (A) / N col (B)
    const int kk   = half * 2;

    const float* Ap = A + l * E_ + kk;                       // A[l, k+kk]
    const float* Wp = W + (size_t)kk * wPitch + colOff + n0 + l; // W[k+kk, col]

    v8f c = {};
#pragma unroll 4
    for (int k = 0; k < E_; k += 4) {
        v2f a, b;
        a[0] = Ap[k];
        a[1] = Ap[k + 1];
        const float* w0 = Wp + (size_t)k * wPitch;
        b[0] = w0[0];
        b[1] = w0[wPitch];
        // D = A x B + C   (args: neg_a, A, neg_b, B, c_mod, C, reuse_a, reuse_b)
        c = __builtin_amdgcn_wmma_f32_16x16x4_f32(false, a, false, b,
                                                  (short)0, c, false, false);
    }

    const float bv = bias[n0 + l];
    // C/D layout: VGPR r -> (M = r + half*8, N = n0 + l)
#pragma unroll
    for (int r = 0; r < 8; ++r) {
        Out[(r + half * 8) * E_ + n0 + l] = c[r] + bv;
    }
}

// Broadcast row[b,:] across all S positions. One float4 per thread, 16 MB write.
__global__ void attn_broadcast(const float4* __restrict__ row4,
                               float4* __restrict__ out4) {
    const int idx = blockIdx.x * blockDim.x + threadIdx.x; // 0..(4*2048*128-1)
    const int b   = idx >> 18;        // / (S_ * E4_) = 262144 = 2^18
    const int e4  = idx & (E4_ - 1);
    out4[idx] = row4[b * E4_ + e4];
}

extern "C" void kernel_launch(void* const* d_in, const int* in_sizes, int n_in,
                              void* d_out, int out_size, void* d_ws, size_t ws_size,
                              hipStream_t stream) {
    const float* x     = (const float*)d_in[0]; // [4,2048,512]
    const float* qkv_w = (const float*)d_in[1]; // [512,1536]
    const float* qkv_b = (const float*)d_in[2]; // [1536]
    const float* out_w = (const float*)d_in[3]; // [512,512]
    const float* out_b = (const float*)d_in[4]; // [512]
    float* out = (float*)d_out;                 // [4,2048,512]

    char* ws = (char*)d_ws;
    // Workspace layout (floats):
    //   partial : SCHUNKS*B_*E_           = 131072 floats (512 KB)
    //   xmeanPad: 16*E_                   =   8192 floats ( 32 KB)
    //   vmeanPad: 16*E_                   =   8192 floats ( 32 KB)
    //   rowPad  : 16*E_                   =   8192 floats ( 32 KB)
    float4* partial  = (float4*)(ws);
    float*  xmeanPad = (float*)(ws + (size_t)SCHUNKS * B_ * E_ * 4);
    float*  vmeanPad = xmeanPad + 16 * E_;
    float*  rowPad   = vmeanPad + 16 * E_;

    // 1) xmean = mean_s(x)  (two-phase deterministic reduction)
    attn_partial_sum<<<B_ * SCHUNKS, E4_, 0, stream>>>((const float4*)x, partial);
    attn_finalize_mean<<<8, 256, 0, stream>>>((const float4*)partial,
                                              (float4*)xmeanPad);

    // 2) vmean = xmean @ Wv + bv   (Wv = qkv_w columns [2E, 3E))
    attn_gemm16_wmma<<<E_ / 16, 32, 0, stream>>>(
        xmeanPad, qkv_w, 3 * E_, 2 * E_, qkv_b + 2 * E_, vmeanPad);

    // 3) row = vmean @ out_w + out_b
    attn_gemm16_wmma<<<E_ / 16, 32, 0, stream>>>(
        vmeanPad, out_w, E_, 0, out_b, rowPad);

    // 4) out[b,s,:] = row[b,:]
    attn_broadcast<<<(B_ * S_ * E4_) / 256, 256, 0, stream>>>(
        (const float4*)rowPad, (float4*)out);
}